// ModelNew_25056839205072
// MI455X (gfx1250) — compile-verified
//
#include <hip/hip_runtime.h>

// out[n,c,p] = x[n,c,p] + alpha * sum_q attn[n,p,q] * D[n,q,c]
// N=2048, C=512, H=W=7 (HW=49)

typedef __attribute__((ext_vector_type(2))) float v2f;
typedef __attribute__((ext_vector_type(4))) float v4f;
typedef __attribute__((ext_vector_type(8))) float v8f;

#define HW_     49
#define CDIM    512
#define KPAD    52      // 13 * 4   (K padded, zeros in LDS)
#define MPAD    64      // 4 * 16   (M padded, zeros in LDS)
#define TSTRIDE 68      // transpose-buffer row stride (16B-aligned rows)

__global__ __launch_bounds__(256)
void attn_bmm_tr_residual(const float* __restrict__ x,
                          const float* __restrict__ attn,
                          const float* __restrict__ D,
                          const float* __restrict__ alpha,
                          float* __restrict__ out)
{
    __shared__ float lds_attn[MPAD * KPAD];     // attn[n] as [p][q], zero padded
    __shared__ float lds_t[8][16 * TSTRIDE];    // per-wave output transpose buffer

    const int n    = blockIdx.x;
    const int tid  = threadIdx.x;
    const int lane = tid & 31;
    const int wave = tid >> 5;
    const int half = lane >> 4;   // 0: lanes 0-15, 1: lanes 16-31
    const int l16  = lane & 15;

    const float a_scale = alpha[0];

    // ---- stage attn[n] (49x49) into LDS as 64x52, zero padded ----
    const size_t attn_base = (size_t)n * HW_ * HW_;
    #pragma unroll
    for (int i = 0; i < (MPAD * KPAD) / 256; ++i) {   // 13 iterations
        int idx = tid + i * 256;
        int row = idx / KPAD;
        int col = idx - row * KPAD;
        float v = 0.0f;
        if (row < HW_ && col < HW_)
            v = attn[attn_base + (size_t)row * HW_ + col];
        lds_attn[idx] = v;
    }
    __syncthreads();

    const size_t d_base = (size_t)n * HW_ * CDIM;   // D[n] : [49][512]
    const size_t o_base = (size_t)n * CDIM * HW_;   // out/x[n] : [512][49]
    float* tbuf = &lds_t[wave][0];

    const bool hi_ok = (lane + 32) < HW_;           // loop-invariant epilogue predicate

    // each wave owns c-tiles: wave, wave+8, wave+16, wave+24
    for (int ct = wave; ct < CDIM / 16; ct += 8) {
        const int cbase = ct * 16;

        // prefetch this wave's x region (16 rows * 49 floats, contiguous 3136B)
        __builtin_prefetch(&x[o_base + (size_t)cbase * HW_ + (size_t)lane * 32], 0, 3);

        v8f acc0 = {}, acc1 = {}, acc2 = {}, acc3 = {};

        // per-lane B base pointer: row (2*half), column cbase+l16.
        // All main-loop B loads are immediate offsets off this one address.
        const float* dptr = D + d_base + (size_t)(2 * half) * CDIM + cbase + l16;
        const int    abase = l16 * KPAD + 2 * half;

        // ---- main K loop: ks = 0..11, rows q = 4*ks + 2*half + {0,1} <= 47, no clamps ----
        #pragma unroll
        for (int ks = 0; ks < 12; ++ks) {
            v2f b;
            b.x = dptr[ks * 4 * CDIM];          // row q,   imm offset 8192*ks
            b.y = dptr[ks * 4 * CDIM + CDIM];   // row q+1, imm offset 8192*ks + 2048

            const int aoff = abase + ks * 4;
            v2f a0 = *(const v2f*)&lds_attn[aoff];
            v2f a1 = *(const v2f*)&lds_attn[aoff + 16 * KPAD];
            v2f a2 = *(const v2f*)&lds_attn[aoff + 32 * KPAD];
            v2f a3 = *(const v2f*)&lds_attn[aoff + 48 * KPAD];

            acc0 = __builtin_amdgcn_wmma_f32_16x16x4_f32(false, a0, false, b, (short)0, acc0, false, false);
            acc1 = __builtin_amdgcn_wmma_f32_16x16x4_f32(false, a1, false, b, (short)0, acc1, false, false);
            acc2 = __builtin_amdgcn_wmma_f32_16x16x4_f32(false, a2, false, b, (short)0, acc2, false, false);
            acc3 = __builtin_amdgcn_wmma_f32_16x16x4_f32(false, a3, false, b, (short)0, acc3, false, false);
        }

        // ---- tail: ks = 12, only q = 48 is real; A is zero for q = 49..51 ----
        {
            float d48 = dptr[48 * CDIM - (size_t)(2 * half) * CDIM];  // row 48 for all lanes
            v2f b; b.x = d48; b.y = d48;                              // A zeros kill invalid k

            const int aoff = abase + 48;
            v2f a0 = *(const v2f*)&lds_attn[aoff];
            v2f a1 = *(const v2f*)&lds_attn[aoff + 16 * KPAD];
            v2f a2 = *(const v2f*)&lds_attn[aoff + 32 * KPAD];
            v2f a3 = *(const v2f*)&lds_attn[aoff + 48 * KPAD];

            acc0 = __builtin_amdgcn_wmma_f32_16x16x4_f32(false, a0, false, b, (short)0, acc0, false, false);
            acc1 = __builtin_amdgcn_wmma_f32_16x16x4_f32(false, a1, false, b, (short)0, acc1, false, false);
            acc2 = __builtin_amdgcn_wmma_f32_16x16x4_f32(false, a2, false, b, (short)0, acc2, false, false);
            acc3 = __builtin_amdgcn_wmma_f32_16x16x4_f32(false, a3, false, b, (short)0, acc3, false, false);
        }

        // ---- transpose through per-wave LDS buffer ----
        // C/D layout: VGPR i, lane L -> p = mt*16 + i + 8*half, c_local = l16
        {
            const int tb = l16 * TSTRIDE + 8 * half;
            *(v4f*)&tbuf[tb +  0    ] = acc0.lo;
            *(v4f*)&tbuf[tb +  0 + 4] = acc0.hi;
            *(v4f*)&tbuf[tb + 16    ] = acc1.lo;
            *(v4f*)&tbuf[tb + 16 + 4] = acc1.hi;
            *(v4f*)&tbuf[tb + 32    ] = acc2.lo;
            *(v4f*)&tbuf[tb + 32 + 4] = acc2.hi;
            *(v4f*)&tbuf[tb + 48    ] = acc3.lo;
            *(v4f*)&tbuf[tb + 48 + 4] = acc3.hi;
        }
        // same-wave LDS RAW: compiler inserts s_wait_dscnt; buffer is wave-private.

        // ---- coalesced residual add + store (streaming, single-use data) ----
        #pragma unroll 4
        for (int r = 0; r < 16; ++r) {
            const size_t g = o_base + (size_t)(cbase + r) * HW_ + lane;
            float v  = tbuf[r * TSTRIDE + lane];
            float xv = __builtin_nontemporal_load(&x[g]);
            __builtin_nontemporal_store(xv + a_scale * v, &out[g]);
        }
        if (hi_ok) {                                  // single divergence region
            #pragma unroll 4
            for (int r = 0; r < 16; ++r) {
                const size_t g = o_base + (size_t)(cbase + r) * HW_ + lane + 32;
                float v  = tbuf[r * TSTRIDE + lane + 32];
                float xv = __builtin_nontemporal_load(&x[g]);
                __builtin_nontemporal_store(xv + a_scale * v, &out[g]);
            }
        }
    }
}

extern "C" void kernel_launch(void* const* d_in, const int* in_sizes, int n_in,
                              void* d_out, int out_size, void* d_ws, size_t ws_size,
                              hipStream_t stream) {
    const float* x     = (const float*)d_in[0];
    const float* attn  = (const float*)d_in[1];
    const float* D     = (const float*)d_in[2];
    const float* alpha = (const float*)d_in[3];
    float* out = (float*)d_out;

    const int N = in_sizes[1] / (HW_ * HW_);   // 2048
    dim3 grid(N), block(256);
    attn_bmm_tr_residual<<<grid, block, 0, stream>>>(x, attn, D, alpha, out);
}